// NodeRegGraphSAGE_7533372637725
// MI455X (gfx1250) — compile-verified
//
#include <hip/hip_runtime.h>
#include <hip/hip_bf16.h>
#include <stdint.h>

// ---------------------------------------------------------------------------
// NodeRegGraphSAGE for MI455X (gfx1250, wave32, WMMA).
// Pipeline: [min-aggregate via ordered-uint atomics] -> [dual-GEMM fused with
// bias + BN-stat accumulation using V_WMMA_F32_16X16X4_F32] -> [BN+ReLU] x2
// -> [final 128->1 linear with wave32 shuffle reduction].
// ---------------------------------------------------------------------------

#define N_NODES 100000
#define N_EDGES 1600000
#define HDIM    128       // H1 == H2 == 128

typedef __attribute__((ext_vector_type(2))) float v2f;
typedef __attribute__((ext_vector_type(8))) float v8f;

// Order-preserving float <-> uint mapping so float-min == unsigned-int-min.
__device__ __forceinline__ unsigned fmin_key(float f) {
    unsigned u = __float_as_uint(f);
    return (u & 0x80000000u) ? ~u : (u | 0x80000000u);
}
__device__ __forceinline__ float fmin_unkey(unsigned k) {
    unsigned u = (k & 0x80000000u) ? (k & 0x7FFFFFFFu) : ~k;
    return __uint_as_float(u);
}

// ---------------------------------------------------------------------------
__global__ void fill_u32_kernel(unsigned* __restrict__ p, unsigned val, long long n) {
    long long i = (long long)blockIdx.x * blockDim.x + threadIdx.x;
    if (i < n) p[i] = val;
}

// One thread per (edge, feature). Gather src feature, atomic-min onto dst.
// f == 0 thread also accumulates in-degree (only when do_deg != 0).
__global__ void edge_min_kernel(const float* __restrict__ feat,
                                const long long* __restrict__ ei, // [2, E]
                                unsigned* __restrict__ keys,      // [N, C]
                                unsigned* __restrict__ deg,       // [N]
                                int nedges, int c_log2, int do_deg) {
    long long idx = (long long)blockIdx.x * blockDim.x + threadIdx.x;
    const int C = 1 << c_log2;
    long long total = (long long)nedges << c_log2;
    if (idx >= total) return;
    int e = (int)(idx >> c_log2);
    int f = (int)(idx & (C - 1));
    int src = (int)ei[e];
    int dst = (int)ei[(long long)nedges + e];
    float v = feat[(long long)src * C + f];
    atomicMin(&keys[(long long)dst * C + f], fmin_key(v));
    if (do_deg && f == 0) atomicAdd(&deg[dst], 1u);
}

// Decode keys -> floats in place; isolated nodes (deg==0) get 0.0 (PyG rule).
__global__ void fixup_kernel(float* __restrict__ agg,
                             const unsigned* __restrict__ deg,
                             long long total, int c_log2) {
    long long i = (long long)blockIdx.x * blockDim.x + threadIdx.x;
    if (i >= total) return;
    int node = (int)(i >> c_log2);
    unsigned k = ((const unsigned*)agg)[i];
    agg[i] = (deg[node] > 0u) ? fmin_unkey(k) : 0.0f;
}

// ---------------------------------------------------------------------------
// Fused dual GEMM:  out = A1 @ W1 + A2 @ W2 + bias   (all fp32, K in {64,128})
// plus per-column sum / sum-of-squares accumulation for BatchNorm.
// Block = 256 threads = 8 waves; wave w owns output columns [16w, 16w+16).
// Block b owns output rows [16b, 16b+16).  Weights staged in LDS.
// wave32 WMMA f32 16x16x4 layouts (ISA 7.12.2):
//   A frag : lane l -> A[m = l&15][k0 = 2*(l>>4) + {0,1}]           (v2f)
//   B frag : lane l -> B[k0 + {0,1}][n = l&15]                      (v2f)
//   C/D    : lane l, vgpr v -> D[m = v + 8*(l>>4)][n = l&15]        (v8f)
__global__ void sage_gemm_kernel(const float* __restrict__ A1,
                                 const float* __restrict__ W1,
                                 const float* __restrict__ A2,
                                 const float* __restrict__ W2,
                                 const float* __restrict__ bias,
                                 float* __restrict__ out,
                                 float* __restrict__ colsum,
                                 float* __restrict__ colsumsq,
                                 int K) {
    extern __shared__ float sW[];          // [2][K * HDIM]
    float* sW1 = sW;
    float* sW2 = sW + (size_t)K * HDIM;

    const int t = threadIdx.x;
    for (int i = t; i < K * HDIM; i += blockDim.x) {
        sW1[i] = W1[i];
        sW2[i] = W2[i];
    }
    __syncthreads();

    const int lane = t & 31;
    const int wave = t >> 5;               // 0..7 -> column tile
    const int n    = lane & 15;            // A row index == B col index
    const int half = lane >> 4;            // 0/1
    const int col  = wave * 16 + n;
    const long long row_base = (long long)blockIdx.x * 16;

    const float* a1p = A1 + (row_base + n) * K + 2 * half;
    const float* a2p = A2 + (row_base + n) * K + 2 * half;

    v8f acc = {};
    for (int kk = 0; kk < K; kk += 4) {
        v2f a, b;
        a.x = a1p[kk];
        a.y = a1p[kk + 1];
        b.x = sW1[(kk + 2 * half) * HDIM + col];
        b.y = sW1[(kk + 2 * half + 1) * HDIM + col];
        acc = __builtin_amdgcn_wmma_f32_16x16x4_f32(false, a, false, b,
                                                    (short)0, acc, false, false);
        a.x = a2p[kk];
        a.y = a2p[kk + 1];
        b.x = sW2[(kk + 2 * half) * HDIM + col];
        b.y = sW2[(kk + 2 * half + 1) * HDIM + col];
        acc = __builtin_amdgcn_wmma_f32_16x16x4_f32(false, a, false, b,
                                                    (short)0, acc, false, false);
    }

    const float bcol = bias[col];
    float s = 0.0f, ss = 0.0f;
#pragma unroll
    for (int v = 0; v < 8; ++v) {
        float val = acc[v] + bcol;
        long long m = row_base + v + 8 * half;
        out[m * HDIM + col] = val;
        s  += val;
        ss += val * val;
    }
    atomicAdd(&colsum[col], s);
    atomicAdd(&colsumsq[col], ss);
}

// mu/var -> per-column affine (scale, shift) for BN in training-normalization
// mode (biased variance), folded with gamma/beta.
__global__ void bn_finalize_kernel(const float* __restrict__ colsum,
                                   const float* __restrict__ colsumsq,
                                   const float* __restrict__ g,
                                   const float* __restrict__ beta,
                                   float* __restrict__ scale,
                                   float* __restrict__ shift,
                                   float invN) {
    int c = blockIdx.x * blockDim.x + threadIdx.x;
    if (c >= HDIM) return;
    float mu  = colsum[c] * invN;
    float var = colsumsq[c] * invN - mu * mu;
    float sc  = g[c] * rsqrtf(var + 1e-5f);
    scale[c] = sc;
    shift[c] = beta[c] - mu * sc;
}

__global__ void bn_apply_relu_kernel(float* __restrict__ h,
                                     const float* __restrict__ scale,
                                     const float* __restrict__ shift,
                                     long long total) {
    long long i = (long long)blockIdx.x * blockDim.x + threadIdx.x;
    if (i >= total) return;
    int c = (int)(i & (HDIM - 1));
    float v = h[i] * scale[c] + shift[c];
    h[i] = v > 0.0f ? v : 0.0f;
}

// One wave32 per node: out[n] = dot(h[n,:128], wf) + bf.
__global__ void final_linear_kernel(const float* __restrict__ h,
                                    const float* __restrict__ wf,
                                    const float* __restrict__ bf,
                                    float* __restrict__ out,
                                    int n_nodes) {
    int node = (int)((long long)blockIdx.x * blockDim.x + threadIdx.x) >> 5;
    int lane = threadIdx.x & 31;
    if (node >= n_nodes) return;
    const float* row = h + (long long)node * HDIM;
    float s = 0.0f;
#pragma unroll
    for (int k = lane; k < HDIM; k += 32) s += row[k] * wf[k];
#pragma unroll
    for (int off = 16; off > 0; off >>= 1) s += __shfl_xor(s, off, 32);
    if (lane == 0) out[node] = s + bf[0];
}

// ---------------------------------------------------------------------------
extern "C" void kernel_launch(void* const* d_in, const int* in_sizes, int n_in,
                              void* d_out, int out_size, void* d_ws, size_t ws_size,
                              hipStream_t stream) {
    (void)in_sizes; (void)n_in; (void)out_size; (void)ws_size;

    const float*     x   = (const float*)d_in[0];
    const long long* ei  = (const long long*)d_in[1];   // int64 [2, E]
    const float*     w1l = (const float*)d_in[2];
    const float*     b1l = (const float*)d_in[3];
    const float*     w1r = (const float*)d_in[4];
    const float*     g1  = (const float*)d_in[5];
    const float*     be1 = (const float*)d_in[6];
    const float*     w2l = (const float*)d_in[7];
    const float*     b2l = (const float*)d_in[8];
    const float*     w2r = (const float*)d_in[9];
    const float*     g2  = (const float*)d_in[10];
    const float*     be2 = (const float*)d_in[11];
    const float*     wf  = (const float*)d_in[12];
    const float*     bf  = (const float*)d_in[13];
    float*           out = (float*)d_out;

    // Workspace layout (~154 MB):
    float*    agg  = (float*)d_ws;                        // N*128 f32 (keys then floats)
    float*    h1   = agg + (size_t)N_NODES * HDIM;        // N*128 f32
    float*    h2   = h1  + (size_t)N_NODES * HDIM;        // N*128 f32
    unsigned* deg  = (unsigned*)(h2 + (size_t)N_NODES * HDIM); // N u32
    float*    st   = (float*)(deg + N_NODES);
    float* colsum = st, * colsumsq = st + HDIM, * scale = st + 2*HDIM, * shift = st + 3*HDIM;

    const int BLK = 256;
    auto nblk = [](long long n, int b) { return (unsigned)((n + b - 1) / b); };

    // ---------------- Layer 1 (K = 64) ----------------
    {
        const int K = 64;
        long long keyN = (long long)N_NODES * K;
        fill_u32_kernel<<<nblk(keyN, BLK), BLK, 0, stream>>>((unsigned*)agg, 0xFFFFFFFFu, keyN);
        fill_u32_kernel<<<nblk(N_NODES, BLK), BLK, 0, stream>>>(deg, 0u, N_NODES);
        fill_u32_kernel<<<1, BLK, 0, stream>>>((unsigned*)colsum, 0u, 2 * HDIM);

        long long work = (long long)N_EDGES * K;
        edge_min_kernel<<<nblk(work, BLK), BLK, 0, stream>>>(x, ei, (unsigned*)agg, deg,
                                                             N_EDGES, 6, 1);
        fixup_kernel<<<nblk(keyN, BLK), BLK, 0, stream>>>(agg, deg, keyN, 6);

        size_t shmem = 2ull * K * HDIM * sizeof(float);   // 64 KB
        sage_gemm_kernel<<<N_NODES / 16, BLK, shmem, stream>>>(agg, w1l, x, w1r, b1l,
                                                               h1, colsum, colsumsq, K);
        bn_finalize_kernel<<<1, HDIM, 0, stream>>>(colsum, colsumsq, g1, be1, scale, shift,
                                                   1.0f / (float)N_NODES);
        bn_apply_relu_kernel<<<nblk((long long)N_NODES * HDIM, BLK), BLK, 0, stream>>>(
            h1, scale, shift, (long long)N_NODES * HDIM);
    }

    // ---------------- Layer 2 (K = 128) ----------------
    {
        const int K = 128;
        long long keyN = (long long)N_NODES * K;
        fill_u32_kernel<<<nblk(keyN, BLK), BLK, 0, stream>>>((unsigned*)agg, 0xFFFFFFFFu, keyN);
        fill_u32_kernel<<<1, BLK, 0, stream>>>((unsigned*)colsum, 0u, 2 * HDIM);

        long long work = (long long)N_EDGES * K;
        edge_min_kernel<<<nblk(work, BLK), BLK, 0, stream>>>(h1, ei, (unsigned*)agg, deg,
                                                             N_EDGES, 7, 0);
        fixup_kernel<<<nblk(keyN, BLK), BLK, 0, stream>>>(agg, deg, keyN, 7);

        size_t shmem = 2ull * K * HDIM * sizeof(float);   // 128 KB (WGP has 320 KB)
        sage_gemm_kernel<<<N_NODES / 16, BLK, shmem, stream>>>(agg, w2l, h1, w2r, b2l,
                                                               h2, colsum, colsumsq, K);
        bn_finalize_kernel<<<1, HDIM, 0, stream>>>(colsum, colsumsq, g2, be2, scale, shift,
                                                   1.0f / (float)N_NODES);
        bn_apply_relu_kernel<<<nblk((long long)N_NODES * HDIM, BLK), BLK, 0, stream>>>(
            h2, scale, shift, (long long)N_NODES * HDIM);
    }

    // ---------------- Final linear (128 -> 1) ----------------
    final_linear_kernel<<<nblk((long long)N_NODES * 32, BLK), BLK, 0, stream>>>(
        h2, wf, bf, out, N_NODES);
}